// GeneralizedInvariantsComputer_21758304321657
// MI455X (gfx1250) — compile-verified
//
#include <hip/hip_runtime.h>

// ---------------------------------------------------------------------------
// Generalized invariants:
//   dC   = det(C @ inv(C)^T)                       [N]      (~1.0 numerically)
//   I_k  = trace(C @ L_k) = sum_ij C[i,j]*L[k,j,i] [N,K]
//   J_k  = dC * trace(L_k)                         [N,K]
//   out  = stack([I, J], axis=2)                   [N,K,2]
//
// Bandwidth-bound (388 MB moved, ~0.3 GFLOP; HBM floor ~17 us @ 23.3 TB/s).
// Strategy: coalesced streaming of Lgen/C through the CDNA5 async global->LDS
// engine (per-lane LDS scatter performs the memory-contiguous -> point-major
// transpose), then per-thread 3x3 algebra out of LDS, one contiguous 64B
// store per point. Full tiles take an unguarded fast path with incremental
// address generation; only the single tail block pays the guarded path.
// ---------------------------------------------------------------------------

#define TPB   128          // threads per block (4 waves)
#define PTS   128          // points per block tile (1 point per thread)
#define LPAD  76           // LDS floats per point for Lgen (304 B = 19*16,
                           // 16B aligned; bank stride 76 -> 2-way conflict max)

__global__ __launch_bounds__(TPB)
void gi_invariants_kernel(const float* __restrict__ Lg,   // [N,8,3,3]
                          const float* __restrict__ Cm,   // [N,3,3]
                          float* __restrict__ out,        // [N,8,2]
                          int N)
{
    __shared__ float sL[PTS * LPAD];   // 38912 B
    __shared__ float sC[PTS * 9];      //  4608 B

    const int t    = threadIdx.x;
    const int tile = blockIdx.x * PTS;       // first point of this block
    const int pc   = min(PTS, N - tile);     // points in this tile (tail-safe)

    const unsigned long long gbL =
        (unsigned long long)(const char*)Lg + (unsigned long long)tile * 288ull;
    const unsigned long long gbC =
        (unsigned long long)(const char*)Cm + (unsigned long long)tile * 36ull;
    const unsigned ldsL = (unsigned)(size_t)(&sL[0]);  // low 32 bits = LDS offset
    const unsigned ldsC = (unsigned)(size_t)(&sC[0]);

    if (pc == PTS) {
        // ---------------- fast path: full tile, no guards ----------------
        // Lgen: 2304 B128 chunks; chunk u -> point p=u/18, slot r=u%18.
        // u steps by 128 = 7*18 + 2  =>  ldsOff += 7*304 + 2*16 = 2160,
        // +16 more when r wraps (p extra +1 net of r-18).
        {
            unsigned p = (unsigned)t / 18u;
            unsigned r = (unsigned)t - p * 18u;
            unsigned ldsOff = ldsL + p * (LPAD * 4u) + r * 16u;
            unsigned long long ga = gbL + (unsigned long long)t * 16ull;
#pragma unroll
            for (int i = 0; i < 18; ++i) {
                asm volatile("global_load_async_to_lds_b128 %0, %1, off"
                             :: "v"(ldsOff), "v"(ga) : "memory");
                r      += 2u;
                ldsOff += 2160u;
                if (r >= 18u) { r -= 18u; ldsOff += 16u; }   // cndmask, no branch
                ga     += 2048ull;
            }
        }
        // C: 1152 B32 chunks, identity mapping (contiguous in LDS too).
        {
            unsigned ldsOff = ldsC + (unsigned)t * 4u;
            unsigned long long ga = gbC + (unsigned long long)t * 4ull;
#pragma unroll
            for (int i = 0; i < 9; ++i) {
                asm volatile("global_load_async_to_lds_b32 %0, %1, off"
                             :: "v"(ldsOff), "v"(ga) : "memory");
                ldsOff += (unsigned)(TPB * 4);
                ga     += (unsigned long long)(TPB * 4);
            }
        }
    } else {
        // ---------------- tail path: guarded staging ----------------
        {
            const int total16 = pc * 18;
#pragma unroll
            for (int i = 0; i < 18; ++i) {
                int u = i * TPB + t;
                if (u < total16) {
                    unsigned p = (unsigned)u / 18u;
                    unsigned r = (unsigned)u - p * 18u;
                    unsigned ldsOff = ldsL + p * (LPAD * 4u) + r * 16u;
                    unsigned long long ga = gbL + (unsigned long long)u * 16ull;
                    asm volatile("global_load_async_to_lds_b128 %0, %1, off"
                                 :: "v"(ldsOff), "v"(ga) : "memory");
                }
            }
        }
        {
            const int total4 = pc * 9;
#pragma unroll
            for (int i = 0; i < 9; ++i) {
                int u = i * TPB + t;
                if (u < total4) {
                    unsigned ldsOff = ldsC + (unsigned)u * 4u;
                    unsigned long long ga = gbC + (unsigned long long)u * 4ull;
                    asm volatile("global_load_async_to_lds_b32 %0, %1, off"
                                 :: "v"(ldsOff), "v"(ga) : "memory");
                }
            }
        }
    }

    // Wait for this wave's async copies, then barrier so every wave's LDS
    // writes are visible to all consumers in the workgroup.
    asm volatile("s_wait_asynccnt 0x0" ::: "memory");
    __syncthreads();

    if (t >= pc) return;

    // ---- Per-point math -----------------------------------------------------
    const float* c = &sC[t * 9];
    const float c00 = c[0], c01 = c[1], c02 = c[2];
    const float c10 = c[3], c11 = c[4], c12 = c[5];
    const float c20 = c[6], c21 = c[7], c22 = c[8];

    // Cofactor matrix of C (note inv(C)^T = cof(C) / det(C))
    const float f00 =  (c11*c22 - c12*c21);
    const float f01 = -(c10*c22 - c12*c20);
    const float f02 =  (c10*c21 - c11*c20);
    const float f10 = -(c01*c22 - c02*c21);
    const float f11 =  (c00*c22 - c02*c20);
    const float f12 = -(c00*c21 - c01*c20);
    const float f20 =  (c01*c12 - c02*c11);
    const float f21 = -(c00*c12 - c02*c10);
    const float f22 =  (c00*c11 - c01*c10);

    const float det  = c00*f00 + c01*f01 + c02*f02;
    const float rdet = 1.0f / det;          // C is SPD-shifted -> det > 0

    // invT = cof * rdet
    const float i00 = f00*rdet, i01 = f01*rdet, i02 = f02*rdet;
    const float i10 = f10*rdet, i11 = f11*rdet, i12 = f12*rdet;
    const float i20 = f20*rdet, i21 = f21*rdet, i22 = f22*rdet;

    // M = C @ inv(C)^T
    const float m00 = c00*i00 + c01*i10 + c02*i20;
    const float m01 = c00*i01 + c01*i11 + c02*i21;
    const float m02 = c00*i02 + c01*i12 + c02*i22;
    const float m10 = c10*i00 + c11*i10 + c12*i20;
    const float m11 = c10*i01 + c11*i11 + c12*i21;
    const float m12 = c10*i02 + c11*i12 + c12*i22;
    const float m20 = c20*i00 + c21*i10 + c22*i20;
    const float m21 = c20*i01 + c21*i11 + c22*i21;
    const float m22 = c20*i02 + c21*i12 + c22*i22;

    const float dC = m00*(m11*m22 - m12*m21)
                   - m01*(m10*m22 - m12*m20)
                   + m02*(m10*m21 - m11*m20);

    // I_k = sum_ij C[i,j] * L[j,i] ; J_k = dC * trace(L_k)
    const float* lb = &sL[t * LPAD];
    float4 o[4];
    float* op = (float*)o;                  // 16 contiguous output floats
#pragma unroll
    for (int k = 0; k < 8; ++k) {
        const float* l = lb + k * 9;
        const float l0 = l[0], l1 = l[1], l2 = l[2];
        const float l3 = l[3], l4 = l[4], l5 = l[5];
        const float l6 = l[6], l7 = l[7], l8 = l[8];

        const float I = c00*l0 + c01*l3 + c02*l6
                      + c10*l1 + c11*l4 + c12*l7
                      + c20*l2 + c21*l5 + c22*l8;
        const float tr = l0 + l4 + l8;

        op[2*k    ] = I;
        op[2*k + 1] = dC * tr;
    }

    // 64 B contiguous store per point: out[(tile+t)*16 .. +15]
    float4* og = (float4*)(out + (size_t)(tile + t) * 16u);
    og[0] = o[0]; og[1] = o[1]; og[2] = o[2]; og[3] = o[3];
}

extern "C" void kernel_launch(void* const* d_in, const int* in_sizes, int n_in,
                              void* d_out, int out_size, void* d_ws, size_t ws_size,
                              hipStream_t stream)
{
    const float* Lg  = (const float*)d_in[0];   // Lgen_ [N,8,3,3] fp32
    const float* Cm  = (const float*)d_in[1];   // C_    [N,3,3]   fp32
    float*       out = (float*)d_out;           //       [N,8,2]   fp32

    const int N = in_sizes[1] / 9;              // C_ has 9 floats per point
    const int blocks = (N + PTS - 1) / PTS;

    hipLaunchKernelGGL(gi_invariants_kernel, dim3(blocks), dim3(TPB), 0, stream,
                       Lg, Cm, out, N);
}